// SelfAttn_47863115547248
// MI455X (gfx1250) — compile-verified
//
#include <hip/hip_runtime.h>
#include <math.h>

typedef __attribute__((ext_vector_type(2))) float v2f;
typedef __attribute__((ext_vector_type(8))) float v8f;

#define SN 2048
#define EN 128
#define ON 128
#define DN 16

// ---------------------------------------------------------------------------
// Kernel 0: delta[s,t] = sum_d time[s,t,d] + sum_d dist[s,t,d]
// Pure streaming reduction: 512 MiB in, 16 MiB out. One element per thread,
// 8x b128 loads, fully bandwidth-bound.
// ---------------------------------------------------------------------------
__global__ __launch_bounds__(256)
void delta_kernel(const float* __restrict__ ttime,
                  const float* __restrict__ tdist,
                  float* __restrict__ delta) {
    int idx = blockIdx.x * 256 + threadIdx.x;          // 0 .. SN*SN-1
    const float4* tp = (const float4*)ttime + (size_t)idx * 4;
    const float4* dp = (const float4*)tdist + (size_t)idx * 4;
    float4 a0 = tp[0], a1 = tp[1], a2 = tp[2], a3 = tp[3];
    float4 b0 = dp[0], b1 = dp[1], b2 = dp[2], b3 = dp[3];
    float s = ((a0.x + a0.y) + (a0.z + a0.w)) + ((a1.x + a1.y) + (a1.z + a1.w))
            + ((a2.x + a2.y) + (a2.z + a2.w)) + ((a3.x + a3.y) + (a3.z + a3.w))
            + ((b0.x + b0.y) + (b0.z + b0.w)) + ((b1.x + b1.y) + (b1.z + b1.w))
            + ((b2.x + b2.y) + (b2.z + b2.w)) + ((b3.x + b3.y) + (b3.z + b3.w));
    delta[idx] = s;
}

// ---------------------------------------------------------------------------
// Kernel 1: Q/K/V = X @ W{q,k,v}. One wave computes one 16x16 tile with
// 32 chained V_WMMA_F32_16X16X4_F32 (K = 128). 3072 tiles, 4 waves/block.
// ---------------------------------------------------------------------------
__global__ __launch_bounds__(128)
void qkv_kernel(const float* __restrict__ X,
                const float* __restrict__ Wq,
                const float* __restrict__ Wk,
                const float* __restrict__ Wv,
                float* __restrict__ Qo,
                float* __restrict__ Ko,
                float* __restrict__ Vo) {
    int wave = threadIdx.x >> 5;
    int lane = threadIdx.x & 31;
    int tile = blockIdx.x * 4 + wave;                 // 0 .. 3071
    int mat  = tile >> 10;                            // 0:Q 1:K 2:V
    int rem  = tile & 1023;
    int s0   = (rem >> 3) * 16;
    int o0   = (rem & 7) * 16;
    const float* W   = (mat == 0) ? Wq : (mat == 1 ? Wk : Wv);
    float*       Out = (mat == 0) ? Qo : (mat == 1 ? Ko : Vo);

    int ln = lane & 15, half = lane >> 4;
    v8f c = 0.f;
#pragma unroll
    for (int ch = 0; ch < 32; ++ch) {
        int k = 4 * ch + 2 * half;
        v2f a = *(const v2f*)&X[(s0 + ln) * EN + k];   // A: rows of X
        v2f b;                                         // B: W[k][o0+ln], W[k+1][o0+ln]
        b.x = W[k * ON + o0 + ln];
        b.y = W[(k + 1) * ON + o0 + ln];
        c = __builtin_amdgcn_wmma_f32_16x16x4_f32(false, a, false, b,
                                                  (short)0, c, false, false);
    }
#pragma unroll
    for (int j = 0; j < 8; ++j)
        Out[(s0 + j + 8 * half) * ON + o0 + ln] = c[j];
}

// ---------------------------------------------------------------------------
// Kernel 2: flash attention with bias. One block (8 waves) per 16-row query
// tile; waves split the key dimension, keep private online-softmax state,
// then merge through LDS (two 64-column rounds to stay < 64 KB static LDS).
// ---------------------------------------------------------------------------
__global__ __launch_bounds__(256)
void attn_kernel(const float* __restrict__ Qm,
                 const float* __restrict__ Km,
                 const float* __restrict__ Vm,
                 const float* __restrict__ delta,
                 float* __restrict__ out) {
    __shared__ float lds_p[8 * 256];       // per-wave P-tile transpose scratch
    __shared__ float lds_acc[8 * 1024];    // per-wave 16x64 acc half
    __shared__ float lds_rm[8][16];
    __shared__ float lds_rl[8][16];
    __shared__ float lds_gm[16];
    __shared__ float lds_gl[16];
    __shared__ float lds_scl[8][16];

    int tid  = threadIdx.x;
    int wave = tid >> 5, lane = tid & 31;
    int ln   = lane & 15, half = lane >> 4;
    int s0   = blockIdx.x * 16;

    // Q tile in A-fragment layout: lane%16 = row, K pair = 4c + 2*half
    v2f qf[32];
#pragma unroll
    for (int ch = 0; ch < 32; ++ch)
        qf[ch] = *(const v2f*)&Qm[(s0 + ln) * ON + 4 * ch + 2 * half];

    v8f acc[8];
#pragma unroll
    for (int nt = 0; nt < 8; ++nt) acc[nt] = 0.f;
    float rm[8], rl[8];
#pragma unroll
    for (int j = 0; j < 8; ++j) { rm[j] = -1e30f; rl[j] = 0.f; }

    float* myp = &lds_p[wave * 256];

    for (int tt = wave; tt < SN / 16; tt += 8) {
        int t0 = tt * 16;

        // scores tile = Q_tile @ K_tile^T  (B frag = contiguous b64 from K row)
        v8f st = 0.f;
#pragma unroll
        for (int ch = 0; ch < 32; ++ch) {
            v2f b = *(const v2f*)&Km[(t0 + ln) * ON + 4 * ch + 2 * half];
            st = __builtin_amdgcn_wmma_f32_16x16x4_f32(false, qf[ch], false, b,
                                                       (short)0, st, false, false);
        }
        // add bias
#pragma unroll
        for (int j = 0; j < 8; ++j)
            st[j] += delta[(size_t)(s0 + j + 8 * half) * SN + t0 + ln];

        // online softmax per row (rows j / j+8 live in 16-lane halves)
#pragma unroll
        for (int j = 0; j < 8; ++j) {
            float v  = st[j];
            float mx = v;
#pragma unroll
            for (int m = 1; m < 16; m <<= 1)
                mx = fmaxf(mx, __shfl_xor(mx, m, 32));
            float newm = fmaxf(rm[j], mx);
            float scl  = __expf(rm[j] - newm);
            float p    = __expf(v - newm);
            float rs   = p;
#pragma unroll
            for (int m = 1; m < 16; m <<= 1)
                rs += __shfl_xor(rs, m, 32);
            rl[j] = rl[j] * scl + rs;
            rm[j] = newm;
#pragma unroll
            for (int nt = 0; nt < 8; ++nt) acc[nt][j] *= scl;
            myp[(j + 8 * half) * 16 + ln] = p;          // C-layout -> LDS
        }

        // reload P in A-fragment layout (same-wave DS ops are in order)
        v2f pf[4];
#pragma unroll
        for (int ch = 0; ch < 4; ++ch)
            pf[ch] = *(const v2f*)&myp[ln * 16 + 4 * ch + 2 * half];

        // acc += P_tile @ V_tile  (K = 16 -> 4 wmma per 16-col group)
#pragma unroll
        for (int nt = 0; nt < 8; ++nt) {
#pragma unroll
            for (int ch = 0; ch < 4; ++ch) {
                int k = t0 + 4 * ch + 2 * half;
                v2f b;
                b.x = Vm[k * ON + nt * 16 + ln];
                b.y = Vm[(k + 1) * ON + nt * 16 + ln];
                acc[nt] = __builtin_amdgcn_wmma_f32_16x16x4_f32(
                    false, pf[ch], false, b, (short)0, acc[nt], false, false);
            }
        }
    }

    // publish per-wave softmax stats
    if (ln == 0) {
#pragma unroll
        for (int j = 0; j < 8; ++j) {
            lds_rm[wave][j + 8 * half] = rm[j];
            lds_rl[wave][j + 8 * half] = rl[j];
        }
    }
    __syncthreads();
    if (tid < 16) {
        float gm = -1e30f;
        for (int w = 0; w < 8; ++w) gm = fmaxf(gm, lds_rm[w][tid]);
        float gl = 0.f;
        for (int w = 0; w < 8; ++w) gl += lds_rl[w][tid] * __expf(lds_rm[w][tid] - gm);
        lds_gm[tid] = gm;
        lds_gl[tid] = gl;
    }
    __syncthreads();
    if (tid < 128) {
        int w = tid >> 4, r = tid & 15;
        lds_scl[w][r] = __expf(lds_rm[w][r] - lds_gm[r]);
    }
    __syncthreads();

    // merge in two 64-column rounds to keep static LDS under 64 KB
#pragma unroll
    for (int round = 0; round < 2; ++round) {
#pragma unroll
        for (int nt = 0; nt < 4; ++nt) {
            int ntg = nt + 4 * round;
#pragma unroll
            for (int j = 0; j < 8; ++j)
                lds_acc[wave * 1024 + (j + 8 * half) * 64 + nt * 16 + ln] = acc[ntg][j];
        }
        __syncthreads();
#pragma unroll
        for (int i = 0; i < 4; ++i) {
            int f = tid + 256 * i;
            int r = f >> 6, cc = f & 63;
            float val = 0.f;
#pragma unroll
            for (int w = 0; w < 8; ++w)
                val += lds_acc[w * 1024 + r * 64 + cc] * lds_scl[w][r];
            out[(s0 + r) * ON + cc + 64 * round] = val / lds_gl[r];
        }
        __syncthreads();
    }
}

// ---------------------------------------------------------------------------
extern "C" void kernel_launch(void* const* d_in, const int* in_sizes, int n_in,
                              void* d_out, int out_size, void* d_ws, size_t ws_size,
                              hipStream_t stream) {
    const float* X     = (const float*)d_in[0];   // [S,E]
    const float* ttime = (const float*)d_in[1];   // [S,S,D]
    const float* tdist = (const float*)d_in[2];   // [S,S,D]
    const float* Wq    = (const float*)d_in[3];   // [E,O]
    const float* Wk    = (const float*)d_in[4];
    const float* Wv    = (const float*)d_in[5];
    float* out = (float*)d_out;

    float* ws    = (float*)d_ws;
    float* delta = ws;                               // S*S
    float* Qw    = delta + (size_t)SN * SN;          // S*O
    float* Kw    = Qw + (size_t)SN * ON;
    float* Vw    = Kw + (size_t)SN * ON;

    delta_kernel<<<(SN * SN) / 256, 256, 0, stream>>>(ttime, tdist, delta);
    qkv_kernel<<<3 * (SN / 16) * (ON / 16) / 4, 128, 0, stream>>>(
        X, Wq, Wk, Wv, Qw, Kw, Vw);
    attn_kernel<<<SN / 16, 256, 0, stream>>>(Qw, Kw, Vw, delta, out);
}